// GraphhMLP_output_6305011991076
// MI455X (gfx1250) — compile-verified
//
#include <hip/hip_runtime.h>
#include <math.h>

// ---------------- problem constants (match reference) ----------------
#define Nn     50000
#define Tt     2
#define Cc     768
#define Gg     16
#define Ee     800000
#define Hh     192
#define TH     384          // T*H floats per node
#define Mrows  100000       // N*T rows for GEMMs
#define EPSV   1e-5f

typedef float v2f __attribute__((ext_vector_type(2)));
typedef float v8f __attribute__((ext_vector_type(8)));

__device__ __forceinline__ float gelu_exact(float v) {
    // torch nn.GELU default: exact erf form
    return 0.5f * v * (1.0f + erff(v * 0.7071067811865476f));
}

// ---------------------------------------------------------------------
// WMMA fp32 GEMM: D[M,Nw] = A[M,K] @ B[K,Nw]  (+bias, +gelu optional)
// 128 threads = 4 waves. Block computes 16 rows x 64 cols.
//   grid.x = M/16, grid.y = Nw/64
// A tile staged in LDS with +4-word row pad -> conflict-free wmma reads.
// ---------------------------------------------------------------------
#define GPAD 4
__global__ __launch_bounds__(128)
void wmma_gemm_f32(const float* __restrict__ A, const float* __restrict__ B,
                   float* __restrict__ D, int M, int K, int Nw,
                   const float* __restrict__ bias, int do_gelu)
{
    __shared__ float lds[16 * (768 + GPAD)];   // 49408 B, sized for K<=768
    const int ldk = K + GPAD;
    const int tid = threadIdx.x;
    const int m0  = blockIdx.x * 16;

    // cooperative, coalesced load of the 16xK A tile (K % 4 == 0)
    const int kv   = K >> 2;            // float4 per row
    const int nvec = 16 * kv;
    for (int v = tid; v < nvec; v += 128) {
        const int row = v / kv;
        const int c4  = (v - row * kv) << 2;
        const float4 t = *(const float4*)(A + (size_t)(m0 + row) * K + c4);
        *(float4*)(&lds[row * ldk + c4]) = t;
    }
    __syncthreads();

    const int wave = tid >> 5;
    const int lane = tid & 31;
    const int row  = lane & 15;          // A-fragment row (M)
    const int koff = (lane >> 4) << 1;   // lanes 16-31 hold K+2,K+3
    const int col  = (blockIdx.y * 4 + wave) * 16 + (lane & 15);

    v8f acc = {};
    const float* lA = &lds[row * ldk + koff];
    const float* pB = B + col;
    for (int k = 0; k < K; k += 4) {
        v2f a, b;
        a.x = lA[k];
        a.y = lA[k + 1];
        b.x = pB[(k + koff) * Nw];
        b.y = pB[(k + koff + 1) * Nw];
        // D = A(16x4) * B(4x16) + C  — fp32 matrix pipe
        acc = __builtin_amdgcn_wmma_f32_16x16x4_f32(false, a, false, b,
                                                    (short)0, acc, false, false);
    }

    const float bv = bias ? bias[col] : 0.0f;
    const int mh = (lane >> 4) << 3;     // lanes 16-31 hold M = r+8
#pragma unroll
    for (int r = 0; r < 8; ++r) {
        float v = acc[r] + bv;
        if (do_gelu) v = gelu_exact(v);
        D[(size_t)(m0 + r + mh) * Nw + col] = v;
    }
}

// ---------------- degree / counts ----------------
__global__ void k_deg(const int* __restrict__ ei, float* __restrict__ deg) {
    int e = blockIdx.x * blockDim.x + threadIdx.x;
    if (e < Ee) atomicAdd(&deg[ei[Ee + e]], 1.0f);   // dst row of edge_index
}
__global__ void k_dinv(float* __restrict__ deg) {
    int n = blockIdx.x * blockDim.x + threadIdx.x;
    if (n < Nn) deg[n] = rsqrtf(deg[n] + 1.0f);      // +1 self loop
}
__global__ void k_cnt(const int* __restrict__ batch, float* __restrict__ cnt) {
    int n = blockIdx.x * blockDim.x + threadIdx.x;
    if (n < Nn) atomicAdd(&cnt[batch[n]], 1.0f);
}

// ---------------- GCN edge aggregation: agg[dst] += h[src]*norm --------
__global__ __launch_bounds__(TH)
void k_agg(const float* __restrict__ h, float* __restrict__ agg,
           const int* __restrict__ ei, const float* __restrict__ dinv)
{
    const int e = blockIdx.x;
    const int c = threadIdx.x;
    const int s = ei[e];
    const int d = ei[Ee + e];
    const float nrm = dinv[s] * dinv[d];
    atomicAdd(&agg[(size_t)d * TH + c], h[(size_t)s * TH + c] * nrm);
}

// self-loop term + conv bias: agg += h*dinv^2 + b
__global__ void k_self_bias(float* __restrict__ agg, const float* __restrict__ h,
                            const float* __restrict__ dinv, const float* __restrict__ b)
{
    int i = blockIdx.x * blockDim.x + threadIdx.x;
    if (i >= Nn * TH) return;
    const int n = i / TH;
    const int j = (i % TH) % Hh;
    const float di = dinv[n];
    agg[i] += h[i] * di * di + b[j];
}

// ---------------- GraphNorm stats: per-(graph,channel) sum & sumsq -----
#define NODES_PB 250
__global__ __launch_bounds__(TH)
void k_stats(const float* __restrict__ h, const int* __restrict__ batch,
             float* __restrict__ gsum, float* __restrict__ gsq)
{
    __shared__ float s_sum[Gg * TH];
    __shared__ float s_sq [Gg * TH];
    const int c = threadIdx.x;
#pragma unroll
    for (int g = 0; g < Gg; ++g) { s_sum[g * TH + c] = 0.0f; s_sq[g * TH + c] = 0.0f; }
    __syncthreads();

    const int n0 = blockIdx.x * NODES_PB;
    const int n1 = (n0 + NODES_PB < Nn) ? (n0 + NODES_PB) : Nn;
    for (int n = n0; n < n1; ++n) {
        const int g = batch[n];                       // broadcast load
        const float v = h[(size_t)n * TH + c];        // coalesced
        atomicAdd(&s_sum[g * TH + c], v);
        atomicAdd(&s_sq [g * TH + c], v * v);
    }
    __syncthreads();
#pragma unroll
    for (int g = 0; g < Gg; ++g) {
        atomicAdd(&gsum[g * TH + c], s_sum[g * TH + c]);
        atomicAdd(&gsq [g * TH + c], s_sq [g * TH + c]);
    }
}

// ---------------- GraphNorm normalize + GELU (+ optional residual) -----
// var(h - a*mean) = E[h^2] - (2a - a^2)*mean^2  (single-pass identity)
__global__ void k_norm(const float* __restrict__ h, const int* __restrict__ batch,
                       const float* __restrict__ cnt, const float* __restrict__ gsum,
                       const float* __restrict__ gsq,
                       const float* __restrict__ w, const float* __restrict__ b,
                       const float* __restrict__ alpha,
                       const float* __restrict__ res, float* __restrict__ out)
{
    int i = blockIdx.x * blockDim.x + threadIdx.x;
    if (i >= Nn * TH) return;
    const int n = i / TH;
    const int c = i % TH;
    const int j = c % Hh;
    const int g = batch[n];
    const float ic   = 1.0f / cnt[g];
    const float mean = gsum[g * TH + c] * ic;
    const float msq  = gsq [g * TH + c] * ic;
    const float a    = alpha[j];
    const float var  = msq - (2.0f * a - a * a) * mean * mean;
    const float o    = h[i] - a * mean;
    float y = w[j] * o * rsqrtf(var + EPSV) + b[j];
    y = gelu_exact(y);
    if (res) y += res[i];
    out[i] = y;
}

// ---------------- final tiny projection 192 -> 3 ----------------------
__global__ void k_head2(const float* __restrict__ h, const float* __restrict__ W,
                        const float* __restrict__ bias, float* __restrict__ out)
{
    int r = blockIdx.x * blockDim.x + threadIdx.x;
    if (r >= Mrows) return;
    const float* hr = h + (size_t)r * Hh;
    float a0 = 0.0f, a1 = 0.0f, a2 = 0.0f;
    for (int k = 0; k < Hh; ++k) {
        const float v = hr[k];
        a0 += v * W[k * 3 + 0];
        a1 += v * W[k * 3 + 1];
        a2 += v * W[k * 3 + 2];
    }
    out[r * 3 + 0] = a0 + bias[0];
    out[r * 3 + 1] = a1 + bias[1];
    out[r * 3 + 2] = a2 + bias[2];
}

// ---------------------------------------------------------------------
extern "C" void kernel_launch(void* const* d_in, const int* in_sizes, int n_in,
                              void* d_out, int out_size, void* d_ws, size_t ws_size,
                              hipStream_t stream)
{
    const float* x     = (const float*)d_in[0];
    const int*   batch = (const int*)  d_in[1];
    const int*   ei    = (const int*)  d_in[2];
    const float* W0    = (const float*)d_in[3];
    const float* b0    = (const float*)d_in[4];
    const float* gn0_w = (const float*)d_in[5];
    const float* gn0_b = (const float*)d_in[6];
    const float* gn0_a = (const float*)d_in[7];
    const float* W1    = (const float*)d_in[8];
    const float* b1    = (const float*)d_in[9];
    const float* gn1_w = (const float*)d_in[10];
    const float* gn1_b = (const float*)d_in[11];
    const float* gn1_a = (const float*)d_in[12];
    const float* Wh1   = (const float*)d_in[13];
    const float* bh1   = (const float*)d_in[14];
    const float* Wh2   = (const float*)d_in[15];
    const float* bh2   = (const float*)d_in[16];

    // ---- carve workspace ----
    size_t off = 0;
    auto carve = [&](size_t bytes) -> void* {
        void* p = (char*)d_ws + off;
        off += (bytes + 255) & ~(size_t)255;
        return p;
    };
    const size_t featB = (size_t)Mrows * Hh * sizeof(float);   // 76.8 MB
    float* hA   = (float*)carve(featB);
    float* hB   = (float*)carve(featB);
    float* hC   = (float*)carve(featB);
    float* dinv = (float*)carve((size_t)Nn * sizeof(float));
    float* cnt  = (float*)carve((size_t)Gg * sizeof(float));
    float* gsum = (float*)carve((size_t)Gg * TH * sizeof(float));
    float* gsq  = (float*)carve((size_t)Gg * TH * sizeof(float));
    if (off > ws_size) return;   // workspace too small; bail safely

    const int eltBlocks = (Nn * TH + 255) / 256;
    const dim3 gemmGrid(Mrows / 16, Hh / 64);   // 6250 x 3

    // ---- degrees, counts ----
    hipMemsetAsync(dinv, 0, (size_t)Nn * sizeof(float), stream);
    hipMemsetAsync(cnt,  0, (size_t)Gg * sizeof(float), stream);
    k_deg <<<(Ee + 255) / 256, 256, 0, stream>>>(ei, dinv);
    k_dinv<<<(Nn + 255) / 256, 256, 0, stream>>>(dinv);
    k_cnt <<<(Nn + 255) / 256, 256, 0, stream>>>(batch, cnt);

    // ---- layer 0: GCNConv 768 -> 192 ----
    wmma_gemm_f32<<<gemmGrid, 128, 0, stream>>>(x, W0, hA, Mrows, Cc, Hh, nullptr, 0);
    hipMemsetAsync(hB, 0, featB, stream);
    k_agg<<<Ee, TH, 0, stream>>>(hA, hB, ei, dinv);
    k_self_bias<<<eltBlocks, 256, 0, stream>>>(hB, hA, dinv, b0);
    hipMemsetAsync(gsum, 0, (size_t)Gg * TH * sizeof(float), stream);
    hipMemsetAsync(gsq,  0, (size_t)Gg * TH * sizeof(float), stream);
    k_stats<<<(Nn + NODES_PB - 1) / NODES_PB, TH, 0, stream>>>(hB, batch, gsum, gsq);
    k_norm<<<eltBlocks, 256, 0, stream>>>(hB, batch, cnt, gsum, gsq,
                                          gn0_w, gn0_b, gn0_a, nullptr, hA);

    // ---- layer 1: GCNConv 192 -> 192 with residual ----
    wmma_gemm_f32<<<gemmGrid, 128, 0, stream>>>(hA, W1, hC, Mrows, Hh, Hh, nullptr, 0);
    hipMemsetAsync(hB, 0, featB, stream);
    k_agg<<<Ee, TH, 0, stream>>>(hC, hB, ei, dinv);
    k_self_bias<<<eltBlocks, 256, 0, stream>>>(hB, hC, dinv, b1);
    hipMemsetAsync(gsum, 0, (size_t)Gg * TH * sizeof(float), stream);
    hipMemsetAsync(gsq,  0, (size_t)Gg * TH * sizeof(float), stream);
    k_stats<<<(Nn + NODES_PB - 1) / NODES_PB, TH, 0, stream>>>(hB, batch, gsum, gsq);
    k_norm<<<eltBlocks, 256, 0, stream>>>(hB, batch, cnt, gsum, gsq,
                                          gn1_w, gn1_b, gn1_a, hA, hC);

    // ---- MLP head ----
    wmma_gemm_f32<<<gemmGrid, 128, 0, stream>>>(hC, Wh1, hB, Mrows, Hh, Hh, bh1, 1);
    k_head2<<<(Mrows + 255) / 256, 256, 0, stream>>>(hB, Wh2, bh2, (float*)d_out);
}